// Net_21646635172356
// MI455X (gfx1250) — compile-verified
//
#include <hip/hip_runtime.h>
#include <math.h>

// ---------------------------------------------------------------------------
// 2-layer GCN for MI455X (gfx1250, wave32).
// GEMMs use V_WMMA_F32_16X16X4_F32 (f32 in/out; GEMM1 is HBM-bound reading x
// at 205 MB so low precision buys nothing). B matrix staged in LDS once per
// block and shared by 8 waves (ds_load_2addr). Edge aggregation uses
// global_atomic_add_f32 into an L2-resident (6.4 MB) accumulator.
// All addressing 32-bit (max buffer 205 MB => float index < 2^26).
// ---------------------------------------------------------------------------

typedef __attribute__((ext_vector_type(2))) float v2f;
typedef __attribute__((ext_vector_type(8))) float v8f;

#define IN_F 512
#define HID  16
#define NCLS 7

// ---------------- degree / norm ----------------
__global__ void k_init_deg(float* deg, int n) {
  int i = blockIdx.x * 256 + threadIdx.x;
  if (i < n) deg[i] = 1.0f;  // self-loop contributes 1
}

__global__ void k_deg_accum(const int* __restrict__ dst, float* deg, int e) {
  int i = blockIdx.x * 256 + threadIdx.x;
  if (i < e) atomicAdd(&deg[dst[i]], 1.0f);
}

__global__ void k_dinv(float* deg, int n) {
  int i = blockIdx.x * 256 + threadIdx.x;
  if (i < n) deg[i] = rsqrtf(deg[i]);  // deg >= 1 always (self-loop)
}

// ---------------- GEMM1: h1[n,16] = x[n,512] @ W1[512,16] ----------------
// 256 threads = 8 wave32; each wave computes one 16x16 output tile.
// K-loop: 128 x V_WMMA_F32_16X16X4_F32.
__global__ __launch_bounds__(256) void k_gemm1(const float* __restrict__ x,
                                               const float* __restrict__ W1,
                                               float* __restrict__ h1, int n) {
  __shared__ float w[IN_F * HID];  // 32 KB
  int tid = threadIdx.x;
  for (int i = tid; i < IN_F * HID / 4; i += 256)
    ((float4*)w)[i] = ((const float4*)W1)[i];
  __syncthreads();

  int wave = tid >> 5;
  int lane = tid & 31;
  int half = lane >> 4;   // 0: K pair {0,1}, 1: K pair {2,3}
  int lm   = lane & 15;   // row within tile (A) / column (B, D)

  int tile0 = (blockIdx.x * 8 + wave) * 16;       // wave-uniform
  int row   = tile0 + lm;
  if (row >= n) row = n - 1;          // clamp: EXEC must stay all-1s for WMMA
  const float* xr = x + row * IN_F + half * 2;    // 32-bit offset (< 2^26)

  v8f acc = {0.f, 0.f, 0.f, 0.f, 0.f, 0.f, 0.f, 0.f};
  #pragma unroll 4
  for (int k = 0; k < IN_F; k += 4) {
    v2f a;                             // A 16x4 f32: VGPR0=K0/K2, VGPR1=K1/K3
    a[0] = xr[k];
    a[1] = xr[k + 1];
    int kk = k + half * 2;             // B 4x16: row-striped across lanes
    v2f b;
    b[0] = w[kk * HID + lm];
    b[1] = w[(kk + 1) * HID + lm];
    acc = __builtin_amdgcn_wmma_f32_16x16x4_f32(
        false, a, false, b, (short)0, acc, false, false);
  }

  // D layout: VGPR v -> M = v (lanes 0-15) / v+8 (lanes 16-31), N = lane&15
  int rbase = tile0 + half * 8;
  if (tile0 + 16 <= n) {               // wave-uniform fast path (full tile)
    #pragma unroll
    for (int v = 0; v < 8; ++v)
      h1[(rbase + v) * HID + lm] = acc[v];
  } else {
    #pragma unroll
    for (int v = 0; v < 8; ++v)
      if (rbase + v < n) h1[(rbase + v) * HID + lm] = acc[v];
  }
}

// ---------------- layer-1 aggregation ----------------
__global__ void k_self1(const float* __restrict__ dinv,
                        const float* __restrict__ h1,
                        float* __restrict__ agg, int total) {  // total = n*16
  int i = blockIdx.x * 256 + threadIdx.x;
  if (i >= total) return;
  int node = i >> 4;
  float dv = dinv[node];
  agg[i] = dv * dv * h1[i];
}

__global__ void k_edge1(const int* __restrict__ src, const int* __restrict__ dst,
                        const float* __restrict__ dinv,
                        const float* __restrict__ h1,
                        float* __restrict__ agg, int e) {
  int i = blockIdx.x * 256 + threadIdx.x;
  if (i >= e) return;
  int s = src[i], d = dst[i];
  float wgt = dinv[s] * dinv[d];
  const float4* hs = (const float4*)(h1 + s * HID);
  float* o = agg + d * HID;
  #pragma unroll
  for (int q = 0; q < 4; ++q) {
    float4 v = hs[q];
    atomicAdd(o + q * 4 + 0, wgt * v.x);
    atomicAdd(o + q * 4 + 1, wgt * v.y);
    atomicAdd(o + q * 4 + 2, wgt * v.z);
    atomicAdd(o + q * 4 + 3, wgt * v.w);
  }
}

__global__ void k_relu_bias(float* agg, const float* __restrict__ b1, int total) {
  int i = blockIdx.x * 256 + threadIdx.x;
  if (i >= total) return;
  agg[i] = fmaxf(agg[i] + b1[i & 15], 0.0f);
}

// ---------------- GEMM2: h2[n,7] = a1[n,16] @ W2[16,7] (padded to 16) -------
__global__ __launch_bounds__(256) void k_gemm2(const float* __restrict__ a1,
                                               const float* __restrict__ W2,
                                               float* __restrict__ h2, int n) {
  __shared__ float w[HID * 16];  // W2 zero-padded to 16x16
  int tid = threadIdx.x;
  {
    int kr = tid >> 4, c = tid & 15;
    w[tid] = (c < NCLS) ? W2[kr * NCLS + c] : 0.0f;
  }
  __syncthreads();

  int wave = tid >> 5;
  int lane = tid & 31;
  int half = lane >> 4;
  int lm   = lane & 15;

  int tile0 = (blockIdx.x * 8 + wave) * 16;
  int row   = tile0 + lm;
  if (row >= n) row = n - 1;
  const float* ar = a1 + row * HID + half * 2;

  v8f acc = {0.f, 0.f, 0.f, 0.f, 0.f, 0.f, 0.f, 0.f};
  #pragma unroll
  for (int k = 0; k < HID; k += 4) {
    v2f a;
    a[0] = ar[k];
    a[1] = ar[k + 1];
    int kk = k + half * 2;
    v2f b;
    b[0] = w[kk * 16 + lm];
    b[1] = w[(kk + 1) * 16 + lm];
    acc = __builtin_amdgcn_wmma_f32_16x16x4_f32(
        false, a, false, b, (short)0, acc, false, false);
  }

  int rbase = tile0 + half * 8;
  bool col_ok = (lm < NCLS);
  if (tile0 + 16 <= n) {
    #pragma unroll
    for (int v = 0; v < 8; ++v)
      if (col_ok) h2[(rbase + v) * NCLS + lm] = acc[v];
  } else {
    #pragma unroll
    for (int v = 0; v < 8; ++v)
      if (col_ok && rbase + v < n) h2[(rbase + v) * NCLS + lm] = acc[v];
  }
}

// ---------------- layer-2 aggregation ----------------
__global__ void k_self2(const float* __restrict__ dinv,
                        const float* __restrict__ h2,
                        float* __restrict__ agg, int n) {
  int i = blockIdx.x * 256 + threadIdx.x;
  if (i >= n * NCLS) return;
  int node = i / NCLS;
  float dv = dinv[node];
  agg[i] = dv * dv * h2[i];
}

__global__ void k_edge2(const int* __restrict__ src, const int* __restrict__ dst,
                        const float* __restrict__ dinv,
                        const float* __restrict__ h2,
                        float* __restrict__ agg, int e) {
  int i = blockIdx.x * 256 + threadIdx.x;
  if (i >= e) return;
  int s = src[i], d = dst[i];
  float wgt = dinv[s] * dinv[d];
  const float* hs = h2 + s * NCLS;
  float* o = agg + d * NCLS;
  #pragma unroll
  for (int c = 0; c < NCLS; ++c) atomicAdd(o + c, wgt * hs[c]);
}

// ---------------- bias + log_softmax ----------------
__global__ void k_logsoftmax(const float* __restrict__ agg,
                             const float* __restrict__ b2,
                             float* __restrict__ out, int n) {
  int i = blockIdx.x * 256 + threadIdx.x;
  if (i >= n) return;
  float z[NCLS];
  float m = -INFINITY;
  #pragma unroll
  for (int c = 0; c < NCLS; ++c) {
    z[c] = agg[i * NCLS + c] + b2[c];
    m = fmaxf(m, z[c]);
  }
  float s = 0.0f;
  #pragma unroll
  for (int c = 0; c < NCLS; ++c) s += __expf(z[c] - m);
  float l = __logf(s) + m;
  #pragma unroll
  for (int c = 0; c < NCLS; ++c) out[i * NCLS + c] = z[c] - l;
}

// ---------------------------------------------------------------------------
extern "C" void kernel_launch(void* const* d_in, const int* in_sizes, int n_in,
                              void* d_out, int out_size, void* d_ws, size_t ws_size,
                              hipStream_t stream) {
  (void)n_in; (void)out_size; (void)ws_size;
  const float* x  = (const float*)d_in[0];
  const float* W1 = (const float*)d_in[1];
  const float* b1 = (const float*)d_in[2];
  const float* W2 = (const float*)d_in[3];
  const float* b2 = (const float*)d_in[4];
  const int*   ei = (const int*)d_in[5];

  const int n = in_sizes[0] / IN_F;      // 100000
  const int e = in_sizes[5] / 2;         // 3200000
  const int* src = ei;
  const int* dst = ei + e;

  // workspace layout (floats): dinv | h1 (reused as h2) | agg1 (reused as agg2)
  float* ws   = (float*)d_ws;
  long   off  = ((long)n + 127) / 128 * 128;
  float* dinv = ws;                        // n floats
  float* h1   = ws + off;                  // n*16 floats (later: h2, n*7)
  float* agg  = h1 + (long)n * HID;        // n*16 floats (later: agg2, n*7)
  float* out  = (float*)d_out;

  const int T = 256;
  dim3 blk(T);

  // norm coefficients
  k_init_deg <<<(n + T - 1) / T, blk, 0, stream>>>(dinv, n);
  k_deg_accum<<<(e + T - 1) / T, blk, 0, stream>>>(dst, dinv, e);
  k_dinv     <<<(n + T - 1) / T, blk, 0, stream>>>(dinv, n);

  // layer 1
  k_gemm1    <<<(n + 127) / 128, blk, 0, stream>>>(x, W1, h1, n);
  k_self1    <<<(n * HID + T - 1) / T, blk, 0, stream>>>(dinv, h1, agg, n * HID);
  k_edge1    <<<(e + T - 1) / T, blk, 0, stream>>>(src, dst, dinv, h1, agg, e);
  k_relu_bias<<<(n * HID + T - 1) / T, blk, 0, stream>>>(agg, b1, n * HID);

  // layer 2 (h2 overwrites h1 buffer; agg2 overwrites agg buffer)
  k_gemm2    <<<(n + 127) / 128, blk, 0, stream>>>(agg, W2, h1, n);
  k_self2    <<<(n * NCLS + T - 1) / T, blk, 0, stream>>>(dinv, h1, agg, n);
  k_edge2    <<<(e + T - 1) / T, blk, 0, stream>>>(src, dst, dinv, h1, agg, e);

  // output
  k_logsoftmax<<<(n + T - 1) / T, blk, 0, stream>>>(agg, b2, out, n);
}